// GPT2FFNInputModel_43001212567954
// MI455X (gfx1250) — compile-verified
//
#include <hip/hip_runtime.h>
#include <hip/hip_bf16.h>

typedef __attribute__((ext_vector_type(2))) float v2f;
typedef __attribute__((ext_vector_type(8))) float v8f;

// Problem constants (reference: B=16, S=2048, D=768, F=3072, L=12)
#define BB 16
#define SS 2048
#define DD 768
#define FF 3072
#define LDS_PAD 4

// ds_swizzle_b32 group-of-32 XOR patterns (offset = xor_mask<<10 | and_mask 0x1F)
#define SWZ_XOR(v, imm) \
    __int_as_float(__builtin_amdgcn_ds_swizzle(__float_as_int(v), (imm)))
#define SWAPX1 0x041F
#define SWAPX2 0x081F
#define SWAPX4 0x101F
#define SWAPX8 0x201F

__global__ void zero_acc_kernel(float* __restrict__ acc) {
    if (threadIdx.x < BB) acc[threadIdx.x] = 0.0f;
}

// One workgroup per 16-column tile of F. 128 threads = 4 waves.
// Wave w handles K-chunk [w*192, w*192+192) with v_wmma_f32_16x16x4_f32,
// using two independent accumulators to halve the WMMA RAW chain on C.
__global__ __launch_bounds__(128) void ffn_tile_kernel(
    const float* __restrict__ x,      // [B, S, D]
    const float* __restrict__ W,      // [L, D, F]
    const int*   __restrict__ layer,  // scalar
    const int*   __restrict__ pos,    // [B]
    float*       __restrict__ acc)    // [B] running row sums
{
    __shared__ float xs[BB][DD + LDS_PAD];
    __shared__ int   posS[BB];

    const int tid = threadIdx.x;
    if (tid < BB) posS[tid] = pos[tid];
    __syncthreads();

    // Gather the 16 needed token rows into LDS (zero for invalid positions).
    for (int idx = tid; idx < BB * DD; idx += 128) {
        const int m = idx / DD;
        const int d = idx - m * DD;
        const int p = posS[m];
        float v = 0.0f;
        if (p >= 0 && p < SS)
            v = x[((size_t)m * SS + (size_t)p) * DD + d];
        xs[m][d] = v;
    }
    __syncthreads();

    const float* __restrict__ W5 = W + (size_t)(*layer) * DD * FF;

    const int wave = tid >> 5;
    const int lane = tid & 31;
    const int half = lane >> 4;   // 0: K lo pair / rows 0-7 of C; 1: K hi pair / rows 8-15
    const int l16  = lane & 15;

    const int f0 = blockIdx.x * 16;          // this WG's 16 columns of F
    const int kChunk = DD / 4;               // 192 K per wave
    const int k0 = wave * kChunk;

    const float* __restrict__ wcol = W5 + f0 + l16;  // column base for B frags

    v8f c0 = {};
    v8f c1 = {};
    #pragma unroll 4
    for (int k = k0; k < k0 + kChunk; k += 8) {
        // --- step A: K = [k, k+4) ---
        {
            const int klo = k + 2 * half;
            v2f a, b;
            a.x = xs[l16][klo];
            a.y = xs[l16][klo + 1];
            b.x = wcol[(size_t)klo * FF];
            b.y = wcol[(size_t)(klo + 1) * FF];
            c0 = __builtin_amdgcn_wmma_f32_16x16x4_f32(
                     false, a, false, b, (short)0, c0, false, false);
        }
        // --- step B: K = [k+4, k+8) --- independent accumulator
        {
            const int klo = k + 4 + 2 * half;
            v2f a, b;
            a.x = xs[l16][klo];
            a.y = xs[l16][klo + 1];
            b.x = wcol[(size_t)klo * FF];
            b.y = wcol[(size_t)(klo + 1) * FF];
            c1 = __builtin_amdgcn_wmma_f32_16x16x4_f32(
                     false, a, false, b, (short)0, c1, false, false);
        }
    }

    // Reduce the 16x16 C tile over its 16 columns.
    // C layout: VGPR i -> row i (lanes 0-15, col=l16) / row i+8 (lanes 16-31).
    // XOR swizzles with masks 1,2,4,8 stay inside each 16-lane half.
    float s[8];
    #pragma unroll
    for (int i = 0; i < 8; ++i) {
        float v = c0[i] + c1[i];
        v += SWZ_XOR(v, SWAPX1);
        v += SWZ_XOR(v, SWAPX2);
        v += SWZ_XOR(v, SWAPX4);
        v += SWZ_XOR(v, SWAPX8);
        s[i] = v;
    }
    if (l16 == 0) {
        #pragma unroll
        for (int i = 0; i < 8; ++i)
            unsafeAtomicAdd(&acc[half * 8 + i], s[i]);
    }
}

// Mean over F commutes with the bias add: out = acc/F + mean(b[5]), masked.
__global__ __launch_bounds__(128) void finalize_kernel(
    const float* __restrict__ bvec,   // [L, F]
    const int*   __restrict__ layer,
    const int*   __restrict__ pos,
    const float* __restrict__ acc,
    float*       __restrict__ out)    // [B]
{
    __shared__ float red[128];
    const float* __restrict__ bl = bvec + (size_t)(*layer) * FF;

    float p = 0.0f;
    for (int f = threadIdx.x; f < FF; f += 128) p += bl[f];
    red[threadIdx.x] = p;
    __syncthreads();
    for (int s = 64; s > 0; s >>= 1) {
        if (threadIdx.x < s) red[threadIdx.x] += red[threadIdx.x + s];
        __syncthreads();
    }
    const float bias_mean = red[0] * (1.0f / (float)FF);

    if (threadIdx.x < BB) {
        const int pp = pos[threadIdx.x];
        const bool valid = (pp >= 0) && (pp < SS);
        out[threadIdx.x] =
            valid ? (acc[threadIdx.x] * (1.0f / (float)FF) + bias_mean) : 0.0f;
    }
}

extern "C" void kernel_launch(void* const* d_in, const int* in_sizes, int n_in,
                              void* d_out, int out_size, void* d_ws, size_t ws_size,
                              hipStream_t stream) {
    const float* x     = (const float*)d_in[0];  // ffn_input [B,S,D]
    const float* W     = (const float*)d_in[1];  // [L,D,F]
    const float* bvec  = (const float*)d_in[2];  // [L,F]
    const int*   layer = (const int*)  d_in[3];  // scalar
    const int*   pos   = (const int*)  d_in[4];  // [B]
    float*       out   = (float*)d_out;
    float*       acc   = (float*)d_ws;           // 16 floats of scratch

    zero_acc_kernel<<<1, 32, 0, stream>>>(acc);
    ffn_tile_kernel<<<FF / 16, 128, 0, stream>>>(x, W, layer, pos, acc);
    finalize_kernel<<<1, 128, 0, stream>>>(bvec, layer, pos, acc, out);
}